// CTCCriterion_32452772888631
// MI455X (gfx1250) — compile-verified
//
#include <hip/hip_runtime.h>
#include <hip/hip_bf16.h>

typedef __attribute__((ext_vector_type(2))) float v2f;
typedef __attribute__((ext_vector_type(8))) float v8f;

#define CTC_S 1024
#define CTC_N 32
#define CTC_C 128
#define CTC_L 128
#define CTC_T 257          // 2L+1
#define CTC_TPAD 288       // 32 lanes * 9 states
#define CTC_LO 1e-5f
#define CTC_SKIP -5.0f

// ---------------------------------------------------------------------------
// log_add: faithful translation of the reference piecewise logaddexp.
// Branch-free select so the wave never diverges.
// ---------------------------------------------------------------------------
__device__ __forceinline__ float ctc_log_add(float x, float y) {
    float d   = x - y;
    float big = fmaxf(x, y);
    float dc  = fminf(fmaxf(d, -20.0f), 20.0f);
    float sml = __logf(__expf(dc) + 1.0f) + y;
    return (fabsf(d) > 10.0f) ? big : sml;
}

// ---------------------------------------------------------------------------
// Kernel 1: row sums via V_WMMA_F32_16X16X4_F32 (B = ones).
// rows r = s*N + n are contiguous 128-float slices of the input.
// One wave handles 16 rows: D[i][j] = sum_c max(LO, x[row_i][c]) for every j.
// Stores logsum[r] = log(rowsum[r]).
// Grid is sized exactly (32768 rows / 16 per wave / 8 waves = 256 blocks),
// so EXEC is all-ones at every WMMA as the ISA requires.
// ---------------------------------------------------------------------------
__global__ __launch_bounds__(256) void ctc_rowsum_wmma(
    const float* __restrict__ inp, float* __restrict__ logsum) {
    const int lane = threadIdx.x & 31;
    const int wave = (blockIdx.x * blockDim.x + threadIdx.x) >> 5;
    const int row0 = wave * 16;

    // A-matrix f32 16x4 layout: lanes 0-15 supply K=0,1 ; lanes 16-31 K=2,3.
    const int mrow   = row0 + (lane & 15);
    const int colsel = (lane >> 4) * 2;
    const float* rp  = inp + mrow * CTC_C + colsel;

    v2f b;
    b.x = 1.0f; b.y = 1.0f;          // all-ones B (4x16)
    v8f acc = {};

    #pragma unroll 4
    for (int kc = 0; kc < CTC_C / 4; ++kc) {
        v2f raw = *(const v2f*)(rp + kc * 4);
        v2f a;
        a.x = fmaxf(CTC_LO, raw.x);
        a.y = fmaxf(CTC_LO, raw.y);
        acc = __builtin_amdgcn_wmma_f32_16x16x4_f32(
            /*neg_a=*/false, a, /*neg_b=*/false, b,
            /*c_mod=*/(short)0, acc, /*reuse_a=*/false, /*reuse_b=*/false);
    }

    // C/D layout: VGPR v, lanes 0-15 hold row M=v, lanes 16-31 hold row M=v+8.
    #pragma unroll
    for (int j = 0; j < 8; ++j) {
        if (lane == j)      logsum[row0 + j]     = __logf(acc[j]);
        if (lane == 16 + j) logsum[row0 + 8 + j] = __logf(acc[j]);
    }
}

// ---------------------------------------------------------------------------
// Kernel 2: forward DP. One wave32 per batch element n.
// Lane owns 9 contiguous states t = lane*9 + j (T padded to 288; states
// >= 257 compute garbage that can never flow backward into real states).
// Per step: 9 gathered probs (software-pipelined), one __shfl_up for the
// roll(v,1) carry, 9 log_adds. No barriers, no LDS.
// ---------------------------------------------------------------------------
__global__ __launch_bounds__(32) void ctc_forward_dp(
    const float* __restrict__ inp, const int* __restrict__ labels,
    const float* __restrict__ logsum, float* __restrict__ losses) {
    const int n    = blockIdx.x;
    const int lane = threadIdx.x;

    int   cls[9];
    float v[9];
    #pragma unroll
    for (int j = 0; j < 9; ++j) {
        const int t = lane * 9 + j;
        int c = 0;                                    // blank (class 0)
        if (t < CTC_T && (t & 1))
            c = labels[((t - 1) >> 1) * CTC_N + n] & (CTC_C - 1);
        cls[j] = c;
        v[j]   = CTC_SKIP * (float)t;                 // v0 = SKIP * arange(T)
    }

    // software pipeline: preload step 0
    float x[9], lz;
    {
        const int rb = n * CTC_C;                     // (s=0)*N + n row
        #pragma unroll
        for (int j = 0; j < 9; ++j) x[j] = inp[rb + cls[j]];
        lz = logsum[n];
    }

    for (int s = 0; s < CTC_S; ++s) {
        // issue next step's loads before using this step's data
        const int sn  = (s + 1 < CTC_S) ? (s + 1) : s;
        const int rbn = (sn * CTC_N + n) * CTC_C;
        float xn[9], lzn;
        #pragma unroll
        for (int j = 0; j < 9; ++j) xn[j] = inp[rbn + cls[j]];
        lzn = logsum[sn * CTC_N + n];

        float m[9];
        #pragma unroll
        for (int j = 0; j < 9; ++j)
            m[j] = __logf(fmaxf(CTC_LO, x[j])) - lz;

        // roll(v,1): carry = previous lane's last state; lane 0 gets SKIP*s
        float carry = __shfl_up(v[8], 1, 32);
        float w0    = (lane == 0) ? (CTC_SKIP * (float)s) : carry;

        #pragma unroll
        for (int j = 8; j >= 1; --j)                  // descending: v[j-1] still old
            v[j] = ctc_log_add(v[j] + m[j], v[j - 1] + m[j]);
        v[0] = ctc_log_add(v[0] + m[0], w0 + m[0]);

        #pragma unroll
        for (int j = 0; j < 9; ++j) x[j] = xn[j];
        lz = lzn;
    }

    // states 255, 256 live in lane 28 (28*9 = 252): slots 3 and 4
    if (lane == 28)
        losses[n] = -ctc_log_add(v[4], v[3]);
}

// ---------------------------------------------------------------------------
// Kernel 3: batch mean (N = 32 = one wave)
// ---------------------------------------------------------------------------
__global__ __launch_bounds__(32) void ctc_reduce(
    const float* __restrict__ losses, float* __restrict__ out) {
    float xv = losses[threadIdx.x];
    #pragma unroll
    for (int off = 16; off > 0; off >>= 1)
        xv += __shfl_down(xv, off, 32);
    if (threadIdx.x == 0) out[0] = xv / (float)CTC_N;
}

extern "C" void kernel_launch(void* const* d_in, const int* in_sizes, int n_in,
                              void* d_out, int out_size, void* d_ws, size_t ws_size,
                              hipStream_t stream) {
    const float* inp     = (const float*)d_in[0];   // (S, N, C) f32
    const int*   targets = (const int*)d_in[1];     // (L, N) int
    float* out    = (float*)d_out;                  // scalar
    float* logsum = (float*)d_ws;                   // 32768 floats
    float* losses = logsum + CTC_S * CTC_N;         // 32 floats

    // 32768 rows / 16 rows-per-wave / 8 waves-per-block = 256 blocks
    ctc_rowsum_wmma<<<256, 256, 0, stream>>>(inp, logsum);
    ctc_forward_dp<<<CTC_N, 32, 0, stream>>>(inp, targets, logsum, losses);
    ctc_reduce<<<1, 32, 0, stream>>>(losses, out);
}